// BatchedPeriodicDistance_3058016714765
// MI455X (gfx1250) — compile-verified
//
#include <hip/hip_runtime.h>
#include <hip/hip_bf16.h>
#include <stdint.h>

// ---------------------------------------------------------------------------
// BatchedPeriodicDistance for MI455X (gfx1250, wave32).
//
// d_out layout (float32, concatenated tuple, E = 4M edges):
//   [0      , 2E) edge_index (cast to float)
//   [2E     , 3E) edge_weight
//   [3E     , 6E) edge_vec
//   [6E     , 9E) shifts_idx copy
//
// d_ws layout (float units):
//   WS_POSW : wrapped positions, padded float4 per atom  (N*4 floats)
//   WS_INV  : per-box inverse, rows padded to 4          (128*12 floats)
//   WS_FLAGS: per-box flags (bit0 cell_is_zero, bit1 pbc)(128 ints)
//   WS_TABLE: per-box 27-entry shift table, float4 each  (128*27*4 floats)
// total ~1.66 MB of scratch.
// ---------------------------------------------------------------------------

#define WS_POSW   0
#define WS_INV    400000
#define WS_FLAGS  401536
#define WS_TABLE  401664
#define MAX_TBL   (128 * 27)   // float4 entries

typedef __attribute__((ext_vector_type(2))) float v2f;
typedef __attribute__((ext_vector_type(8))) float v8f;
typedef __attribute__((ext_vector_type(4))) int   v4i;

// ---- CDNA5 async global->LDS copy (ASYNCcnt-tracked) ----------------------
__device__ __forceinline__ void ld_async_b128(float4* lds_dst, const float4* g_src) {
#if defined(__gfx1250__) && __has_builtin(__builtin_amdgcn_global_load_async_to_lds_b128)
    __builtin_amdgcn_global_load_async_to_lds_b128(
        (__attribute__((address_space(1))) v4i*)g_src,
        (__attribute__((address_space(3))) v4i*)lds_dst,
        0 /*imm offset*/, 0 /*cpol*/);
#else
    *lds_dst = *g_src;   // synchronous fallback (global_load_b128 + ds_store_b128)
#endif
}

__device__ __forceinline__ void wait_async0() {
#if defined(__gfx1250__) && __has_builtin(__builtin_amdgcn_s_wait_asynccnt)
    __builtin_amdgcn_s_wait_asynccnt(0);
#elif defined(__gfx1250__)
    asm volatile("s_wait_asynccnt 0" ::: "memory");
#endif
}

// ---------------------------------------------------------------------------
// Kernel A: one wave per box. Computes adjugate inverse (== pinv for these
// non-singular cells), pbc flags, and the 27-entry {-1,0,1}^3 @ box shift
// table. The table is a genuine 27x3 @ 3x3 product -> two
// V_WMMA_F32_16X16X4_F32 with a wave-uniform B operand.
// ---------------------------------------------------------------------------
__global__ void __launch_bounds__(32)
box_prep_kernel(const float* __restrict__ box, float* __restrict__ ws)
{
    const int  b    = blockIdx.x;
    const int  lane = threadIdx.x;   // 0..31
    const int  n    = lane & 15;
    const bool hi   = lane > 15;

    // Uniform address -> scalar loads.
    const float m0 = box[b*9+0], m1 = box[b*9+1], m2 = box[b*9+2];
    const float m3 = box[b*9+3], m4 = box[b*9+4], m5 = box[b*9+5];
    const float m6 = box[b*9+6], m7 = box[b*9+7], m8 = box[b*9+8];

    const bool cz = (m0==0.f)&&(m1==0.f)&&(m2==0.f)&&(m3==0.f)&&(m4==0.f)&&
                    (m5==0.f)&&(m6==0.f)&&(m7==0.f)&&(m8==0.f);

    const float det = m0*(m4*m8 - m5*m7) - m1*(m3*m8 - m5*m6) + m2*(m3*m7 - m4*m6);
    const float rd  = (!cz && det != 0.f) ? 1.f/det : 0.f;
    const float i00=(m4*m8-m5*m7)*rd, i01=(m2*m7-m1*m8)*rd, i02=(m1*m5-m2*m4)*rd;
    const float i10=(m5*m6-m3*m8)*rd, i11=(m0*m8-m2*m6)*rd, i12=(m2*m3-m0*m5)*rd;
    const float i20=(m3*m7-m4*m6)*rd, i21=(m1*m6-m0*m7)*rd, i22=(m0*m4-m1*m3)*rd;

    float* tab = ws + WS_TABLE + (size_t)b * 27 * 4;

#if defined(__gfx1250__) && __has_builtin(__builtin_amdgcn_wmma_f32_16x16x4_f32)
    // --- B operand: 4x16, B[k][col] = box[k][col] (k<3, col<3), else 0.
    // Layout: vgpr0 = {K=0 | K=2}, vgpr1 = {K=1 | K=3}, N = lane%16.
    const int nc = (n < 3) ? n : 0;                // clamp to stay in-bounds
    const float bk02 = box[b*9 + (hi ? 6 : 0) + nc];
    const float bk1  = box[b*9 + 3 + nc];
    v2f B; B.x = (n < 3) ? bk02 : 0.f;
    B.y = (n < 3 && !hi) ? bk1 : 0.f;

    // --- A operands: shift-pattern tiles (rows = shift code 0..26).
    // code = (sx+1)*9 + (sy+1)*3 + (sz+1). Layout: vgpr0={K=0|K=2}, vgpr1={K=1|K=3}.
    const int r0 = n;            // tile 0 rows 0..15
    const int r1 = 16 + n;       // tile 1 rows 16..31 (only <27 used)
    const float sx0 = (float)(r0/9 - 1), sy0 = (float)((r0/3)%3 - 1), sz0 = (float)(r0%3 - 1);
    const float sx1 = (r1 < 27) ? (float)(r1/9 - 1)     : 0.f;
    const float sy1 = (r1 < 27) ? (float)((r1/3)%3 - 1) : 0.f;
    const float sz1 = (r1 < 27) ? (float)(r1%3 - 1)     : 0.f;
    v2f A0; A0.x = hi ? sz0 : sx0;  A0.y = hi ? 0.f : sy0;
    v2f A1; A1.x = hi ? sz1 : sx1;  A1.y = hi ? 0.f : sy1;

    v8f c = {};
    // EXEC is full here (no divergent branches yet) per WMMA restriction.
    v8f d0 = __builtin_amdgcn_wmma_f32_16x16x4_f32(false, A0, false, B, (short)0, c, false, false);
    v8f d1 = __builtin_amdgcn_wmma_f32_16x16x4_f32(false, A1, false, B, (short)0, c, false, false);

    // D layout: vgpr r, lanes0-15 -> row r, lanes16-31 -> row 8+r; col = lane%16.
    // Columns 3..15 are exact zeros (B padded), so col 3 doubles as the pad.
    if (n < 4) {
        #pragma unroll
        for (int r = 0; r < 8; ++r) {
            const int row0 = hi ? 8 + r : r;
            tab[row0 * 4 + n] = d0[r];
            const int row1 = 16 + row0;
            if (row1 < 27) tab[row1 * 4 + n] = d1[r];
        }
    }
#else
    // Scalar fallback: lane L computes table row L.
    if (lane < 27) {
        const float sx = (float)(lane/9 - 1), sy = (float)((lane/3)%3 - 1), sz = (float)(lane%3 - 1);
        tab[lane*4+0] = sx*m0 + sy*m3 + sz*m6;
        tab[lane*4+1] = sx*m1 + sy*m4 + sz*m7;
        tab[lane*4+2] = sx*m2 + sy*m5 + sz*m8;
        tab[lane*4+3] = 0.f;
    }
#endif

    if (lane == 0) {
        float4* ivp = (float4*)(ws + WS_INV + (size_t)b * 12);
        ivp[0] = make_float4(i00, i01, i02, 0.f);
        ivp[1] = make_float4(i10, i11, i12, 0.f);
        ivp[2] = make_float4(i20, i21, i22, 0.f);
        ((int*)(ws + WS_FLAGS))[b] = (cz ? 1 : 0) | (cz ? 0 : 2);  // pbc = !cell_is_zero
    }
}

// ---------------------------------------------------------------------------
// Kernel B: per-atom wrap: frac = p @ inv; frac %= 1 (if pbc); out = frac @ box.
// Result stored padded to float4 so edge gathers are single b128 loads.
// ---------------------------------------------------------------------------
__global__ void __launch_bounds__(256)
wrap_atoms_kernel(const float* __restrict__ pos, const float* __restrict__ box,
                  const int* __restrict__ batch, float* __restrict__ ws, int n_atoms)
{
    const int a = blockIdx.x * 256 + threadIdx.x;
    if (a >= n_atoms) return;
    const int b = batch[a];

    const float px = pos[a*3+0], py = pos[a*3+1], pz = pos[a*3+2];
    const int   fl  = ((const int*)(ws + WS_FLAGS))[b];
    const bool  cz  = (fl & 1) != 0;
    const bool  pbc = (fl & 2) != 0;

    const float4* ivp = (const float4*)(ws + WS_INV + (size_t)b * 12);
    const float4 r0 = ivp[0], r1 = ivp[1], r2 = ivp[2];

    float fx = px*r0.x + py*r1.x + pz*r2.x;
    float fy = px*r0.y + py*r1.y + pz*r2.y;
    float fz = px*r0.z + py*r1.z + pz*r2.z;
    if (cz) { fx = px; fy = py; fz = pz; }
    if (pbc) { fx -= floorf(fx); fy -= floorf(fy); fz -= floorf(fz); }

    const float m0 = box[b*9+0], m1 = box[b*9+1], m2 = box[b*9+2];
    const float m3 = box[b*9+3], m4 = box[b*9+4], m5 = box[b*9+5];
    const float m6 = box[b*9+6], m7 = box[b*9+7], m8 = box[b*9+8];
    float ox = fx*m0 + fy*m3 + fz*m6;
    float oy = fx*m1 + fy*m4 + fz*m7;
    float oz = fx*m2 + fy*m5 + fz*m8;
    if (cz) { ox = fx; oy = fy; oz = fz; }

    ((float4*)(ws + WS_POSW))[a] = make_float4(ox, oy, oz, 0.f);
}

// ---------------------------------------------------------------------------
// Kernel C: per-edge. Persistent grid (512 blocks) so the 54 KB shift table is
// staged into LDS once per block via async global->LDS b128 copies, replacing
// the per-edge 3x3 matvec with one LDS float4 lookup.
// ---------------------------------------------------------------------------
__global__ void __launch_bounds__(256)
edges_kernel(const int* __restrict__ eidx, const float* __restrict__ shifts,
             const int* __restrict__ batch, const float* __restrict__ ws,
             float* __restrict__ out, long E, int n_boxes)
{
    __shared__ float4 s_table[MAX_TBL];           // 55,296 B < 64 KB static cap

    const int tid = threadIdx.x;
    const float4* gtab = (const float4*)(ws + WS_TABLE);
    const int tot = n_boxes * 27;
    for (int idx = tid; idx < tot; idx += 256)
        ld_async_b128(&s_table[idx], &gtab[idx]);
    wait_async0();
    __syncthreads();

    const float4* posw = (const float4*)(ws + WS_POSW);
    float* o_w   = out + 2*E;
    float* o_vec = out + 3*E;
    float* o_sh  = out + 6*E;

    const long stride = (long)gridDim.x * 256;
    for (long e = (long)blockIdx.x * 256 + tid; e < E; e += stride) {
        const int   i  = eidx[e];
        const int   j  = eidx[E + e];
        const float s0 = shifts[e*3+0];
        const float s1 = shifts[e*3+1];
        const float s2 = shifts[e*3+2];
        const int   b  = batch[i];
        const int code = ((int)s0 + 1)*9 + ((int)s1 + 1)*3 + ((int)s2 + 1);

        const float4 cs = s_table[b*27 + code];
        const float4 pi = posw[i];
        const float4 pj = posw[j];

        const float dx = pj.x - pi.x + cs.x;
        const float dy = pj.y - pi.y + cs.y;
        const float dz = pj.z - pi.z + cs.z;

        out[e]       = (float)i;
        out[E + e]   = (float)j;
        o_w[e]       = sqrtf(dx*dx + dy*dy + dz*dz);
        o_vec[e*3+0] = -dx;
        o_vec[e*3+1] = -dy;
        o_vec[e*3+2] = -dz;
        o_sh[e*3+0]  = s0;
        o_sh[e*3+1]  = s1;
        o_sh[e*3+2]  = s2;
    }
}

// ---------------------------------------------------------------------------
extern "C" void kernel_launch(void* const* d_in, const int* in_sizes, int n_in,
                              void* d_out, int out_size, void* d_ws, size_t ws_size,
                              hipStream_t stream)
{
    const float* pos    = (const float*)d_in[0];
    const float* box    = (const float*)d_in[1];
    const int*   batch  = (const int*)d_in[2];
    const int*   eidx   = (const int*)d_in[3];
    const float* shifts = (const float*)d_in[4];
    float*       ws     = (float*)d_ws;
    float*       out    = (float*)d_out;

    const int  n_atoms = in_sizes[0] / 3;
    const int  n_boxes = in_sizes[1] / 9;
    const long E       = (long)in_sizes[3] / 2;

    box_prep_kernel<<<n_boxes, 32, 0, stream>>>(box, ws);
    wrap_atoms_kernel<<<(n_atoms + 255) / 256, 256, 0, stream>>>(pos, box, batch, ws, n_atoms);
    edges_kernel<<<512, 256, 0, stream>>>(eidx, shifts, batch, ws, out, E, n_boxes);
}